// GNNML1PRO_38422777430260
// MI455X (gfx1250) — compile-verified
//
#include <hip/hip_runtime.h>
#include <math.h>

typedef __attribute__((ext_vector_type(2))) float v2f;
typedef __attribute__((ext_vector_type(8))) float v8f;

#define N_NODES   50000
#define N_EDGES   800000
#define D_IN      64
#define NIN       144
#define N_CLASSES 128

// ---------------------------------------------------------------------------
// fp32 WMMA tile: D(16x16) += A(16x4) * B(4x16) over K, via V_WMMA_F32_16X16X4_F32.
// Lane mapping (ISA 7.12.2): idx = lane%16, half = lane/16.
//   A frag (registers): row M=idx, K = kk + 2*half + {0,1}
//   B frag (LDS):       col N=idx, K = kk + 2*half + {0,1}   (pair-interleaved -> b64)
//   C frag:             acc[r] = C[M = r + 8*half][N = idx]
// LDS weight layout: lw[(k>>1)*(2*TOTC) + c*2 + (k&1)] = W[k][c]  (K-pair interleave)
// ---------------------------------------------------------------------------
template <int K, int TOTC>
__device__ __forceinline__ v8f gemm_tile_lds(const v2f* __restrict__ af,
                                             const float* __restrict__ lw,
                                             int col0, int idx, int half, v8f acc) {
#pragma unroll
  for (int kk = 0; kk < K; kk += 4) {
    const int kp = (kk >> 1) + half;                 // pair index of k = kk + 2*half
    const float* bp = lw + (size_t)kp * (2 * TOTC) + (size_t)(col0 + idx) * 2;
    v2f b; b[0] = bp[0]; b[1] = bp[1];               // single ds_load_b64
    acc = __builtin_amdgcn_wmma_f32_16x16x4_f32(false, af[kk >> 2], false, b,
                                                (short)0, acc, false, false);
  }
  return acc;
}

// ---------------------------------------------------------------------------
// Fused per-node transform for one GNNML1 block:
//   h[:,   0: 64] = relu(A @ Wmain + bmain)
//   xw[:,  0: 64] = A @ Wc                 (pre-bias, fed to edge scatter)
//   h[:, 128:144] = relu(A@Wp0+bp0) * relu(A@Wp1+bp1)
// Weights staged into LDS once per workgroup (TOTC = 64+64+16+16 = 160 cols).
// ---------------------------------------------------------------------------
template <int KIN>
__global__ void __launch_bounds__(256)
k_node_gemm(const float* __restrict__ A,
            const float* __restrict__ Wmain, const float* __restrict__ bmain,
            const float* __restrict__ Wc,
            const float* __restrict__ Wp0, const float* __restrict__ bp0,
            const float* __restrict__ Wp1, const float* __restrict__ bp1,
            float* __restrict__ h, float* __restrict__ xw) {
  constexpr int TOTC = 160;
  extern __shared__ float lw[];                      // [KIN/2][2*TOTC] pair-interleaved

  // --- cooperative weight staging (all 256 threads; before any wave exits) ---
  for (int i = threadIdx.x; i < KIN * TOTC; i += 256) {
    const int k = i / TOTC;
    const int c = i % TOTC;
    float v;
    if (c < 64)        v = Wmain[(size_t)k * 64 + c];
    else if (c < 128)  v = Wc  [(size_t)k * 64 + (c - 64)];
    else if (c < 144)  v = Wp0 [(size_t)k * 16 + (c - 128)];
    else               v = Wp1 [(size_t)k * 16 + (c - 144)];
    lw[(size_t)(k >> 1) * (2 * TOTC) + (size_t)c * 2 + (k & 1)] = v;
  }
  __syncthreads();

  const int wave = threadIdx.x >> 5;
  const int lane = threadIdx.x & 31;
  const int idx  = lane & 15;
  const int half = lane >> 4;
  const int tile = blockIdx.x * 8 + wave;
  if (tile >= N_NODES / 16) return;                  // wave-uniform: EXEC all-ones
  const int row0 = tile * 16;

  // --- hoist A fragments: 16 rows x KIN, reused across all 10 column tiles ---
  v2f af[KIN / 4];
  {
    const float* arow = A + (size_t)(row0 + idx) * KIN + 2 * half;
#pragma unroll
    for (int kk = 0; kk < KIN; kk += 4) {            // global_load_b64 each
      af[kk >> 2][0] = arow[kk];
      af[kk >> 2][1] = arow[kk + 1];
    }
  }

#pragma unroll
  for (int t = 0; t < 4; ++t) {                      // 64 cols of Wmain -> h[:,0:64]
    v8f acc = {};
    acc = gemm_tile_lds<KIN, TOTC>(af, lw, t * 16, idx, half, acc);
    const float bb = bmain[t * 16 + idx];
#pragma unroll
    for (int r = 0; r < 8; ++r) {
      const int node = row0 + r + 8 * half;
      const float v = acc[r] + bb;
      h[(size_t)node * NIN + t * 16 + idx] = v > 0.f ? v : 0.f;
    }
  }
#pragma unroll
  for (int t = 0; t < 4; ++t) {                      // 64 cols of Wc -> xw
    v8f acc = {};
    acc = gemm_tile_lds<KIN, TOTC>(af, lw, 64 + t * 16, idx, half, acc);
#pragma unroll
    for (int r = 0; r < 8; ++r) {
      const int node = row0 + r + 8 * half;
      xw[(size_t)node * 64 + t * 16 + idx] = acc[r];
    }
  }
  {                                                  // gated product -> h[:,128:144]
    v8f a0 = {}; a0 = gemm_tile_lds<KIN, TOTC>(af, lw, 128, idx, half, a0);
    v8f a1 = {}; a1 = gemm_tile_lds<KIN, TOTC>(af, lw, 144, idx, half, a1);
    const float c0 = bp0[idx], c1 = bp1[idx];
#pragma unroll
    for (int r = 0; r < 8; ++r) {
      const int node = row0 + r + 8 * half;
      float u = a0[r] + c0; u = u > 0.f ? u : 0.f;
      float v = a1[r] + c1; v = v > 0.f ? v : 0.f;
      h[(size_t)node * NIN + 128 + idx] = u * v;
    }
  }
}

// ---------------------------------------------------------------------------
// agg[dst, c:c+4] += ew[e] * xw[src, c:c+4]  (float4 per thread; 16 threads/edge)
// ---------------------------------------------------------------------------
__global__ void __launch_bounds__(256)
k_scatter(const long long* __restrict__ ei, const float* __restrict__ ea,
          const float* __restrict__ xw, float* __restrict__ agg) {
  const unsigned gid = blockIdx.x * blockDim.x + threadIdx.x;
  if (gid >= (unsigned)N_EDGES * 16u) return;
  const int e  = gid >> 4;
  const int c4 = (gid & 15) << 2;
  const long long s = ei[e];
  const long long d = ei[N_EDGES + e];
  const float w = ea[e];
  const float4 v = *(const float4*)(xw + (size_t)s * 64 + c4);
  float* ap = agg + (size_t)d * 64 + c4;
  atomicAdd(ap + 0, w * v.x);
  atomicAdd(ap + 1, w * v.y);
  atomicAdd(ap + 2, w * v.z);
  atomicAdd(ap + 3, w * v.w);
}

// h[:, 64:128] = relu(agg + bc)
__global__ void __launch_bounds__(256)
k_aggrelu(const float* __restrict__ agg, const float* __restrict__ bc,
          float* __restrict__ h) {
  const unsigned gid = blockIdx.x * blockDim.x + threadIdx.x;
  if (gid >= (unsigned)N_NODES * 64u) return;
  const int n = gid >> 6;
  const int c = gid & 63;
  const float v = agg[gid] + bc[c];
  h[(size_t)n * NIN + 64 + c] = v > 0.f ? v : 0.f;
}

// ---------------------------------------------------------------------------
// logits = h2 @ Wf + bf  (WMMA, K=144, 8 col tiles; Wf staged in LDS)
// ---------------------------------------------------------------------------
__global__ void __launch_bounds__(256)
k_classifier(const float* __restrict__ h, const float* __restrict__ Wf,
             const float* __restrict__ bf, float* __restrict__ out) {
  constexpr int TOTC = N_CLASSES;                    // 128
  extern __shared__ float lw[];                      // [NIN/2][256] pair-interleaved

  for (int i = threadIdx.x; i < NIN * TOTC; i += 256) {
    const int k = i / TOTC;
    const int c = i % TOTC;
    lw[(size_t)(k >> 1) * (2 * TOTC) + (size_t)c * 2 + (k & 1)] =
        Wf[(size_t)k * TOTC + c];
  }
  __syncthreads();

  const int wave = threadIdx.x >> 5;
  const int lane = threadIdx.x & 31;
  const int idx  = lane & 15;
  const int half = lane >> 4;
  const int tile = blockIdx.x * 8 + wave;
  if (tile >= N_NODES / 16) return;
  const int row0 = tile * 16;

  v2f af[NIN / 4];
  {
    const float* arow = h + (size_t)(row0 + idx) * NIN + 2 * half;
#pragma unroll
    for (int kk = 0; kk < NIN; kk += 4) {
      af[kk >> 2][0] = arow[kk];
      af[kk >> 2][1] = arow[kk + 1];
    }
  }

#pragma unroll
  for (int t = 0; t < 8; ++t) {
    v8f acc = {};
    acc = gemm_tile_lds<NIN, TOTC>(af, lw, t * 16, idx, half, acc);
    const float bb = bf[t * 16 + idx];
#pragma unroll
    for (int r = 0; r < 8; ++r) {
      const int node = row0 + r + 8 * half;
      out[(size_t)node * N_CLASSES + t * 16 + idx] = acc[r] + bb;
    }
  }
}

// In-place row-wise log_softmax, one 128-thread block per node.
__global__ void __launch_bounds__(128)
k_logsoftmax(float* __restrict__ out) {
  const int n = blockIdx.x;
  const int t = threadIdx.x;
  __shared__ float red[N_CLASSES];
  const float v = out[(size_t)n * N_CLASSES + t];
  red[t] = v;
  __syncthreads();
  for (int s = N_CLASSES / 2; s > 0; s >>= 1) {
    if (t < s) red[t] = fmaxf(red[t], red[t + s]);
    __syncthreads();
  }
  const float m = red[0];
  __syncthreads();
  red[t] = __expf(v - m);
  __syncthreads();
  for (int s = N_CLASSES / 2; s > 0; s >>= 1) {
    if (t < s) red[t] += red[t + s];
    __syncthreads();
  }
  const float lse = __logf(red[0]);
  out[(size_t)n * N_CLASSES + t] = v - m - lse;
}

extern "C" void kernel_launch(void* const* d_in, const int* in_sizes, int n_in,
                              void* d_out, int out_size, void* d_ws, size_t ws_size,
                              hipStream_t stream) {
  (void)in_sizes; (void)n_in; (void)out_size; (void)ws_size;

  const float*     x   = (const float*)d_in[0];
  const long long* ei  = (const long long*)d_in[1];   // int64 [2, E]
  const float*     ea  = (const float*)d_in[2];
  const float*     Wc1 = (const float*)d_in[3];       // [1,64,64]
  const float*     bc1 = (const float*)d_in[4];
  const float*     Wc2 = (const float*)d_in[5];       // [1,144,64]
  const float*     bc2 = (const float*)d_in[6];
  const float*     W11 = (const float*)d_in[7];
  const float*     b11 = (const float*)d_in[8];
  const float*     W12 = (const float*)d_in[9];
  const float*     b12 = (const float*)d_in[10];
  const float*     W13 = (const float*)d_in[11];
  const float*     b13 = (const float*)d_in[12];
  const float*     W21 = (const float*)d_in[13];
  const float*     b21 = (const float*)d_in[14];
  const float*     W22 = (const float*)d_in[15];
  const float*     b22 = (const float*)d_in[16];
  const float*     W23 = (const float*)d_in[17];
  const float*     b23 = (const float*)d_in[18];
  const float*     Wf  = (const float*)d_in[19];
  const float*     bf  = (const float*)d_in[20];

  float* out = (float*)d_out;
  float* xw  = (float*)d_ws;                          // [N, 64]  (reused both blocks)
  float* agg = xw  + (size_t)N_NODES * 64;            // [N, 64]  (reused both blocks)
  float* h1  = agg + (size_t)N_NODES * 64;            // [N, 144]
  float* h2  = h1  + (size_t)N_NODES * NIN;           // [N, 144]

  const int tiles   = N_NODES / 16;                   // 3125 (exact)
  const int gBlocks = (tiles + 7) / 8;                // 8 waves/block
  const dim3 b256(256);
  const unsigned edgeThreads = (unsigned)N_EDGES * 16u;
  const unsigned nodeThreads = (unsigned)N_NODES * 64u;

  const size_t lds1 = (size_t)D_IN * 160 * sizeof(float);       // 40 KB
  const size_t lds2 = (size_t)NIN  * 160 * sizeof(float);       // 90 KB
  const size_t ldsC = (size_t)NIN  * N_CLASSES * sizeof(float); // 72 KB

  // ---- Block 1 ----
  hipMemsetAsync(agg, 0, (size_t)N_NODES * 64 * sizeof(float), stream);
  k_node_gemm<D_IN><<<gBlocks, b256, lds1, stream>>>(x, W11, b11, Wc1, W12, b12,
                                                     W13, b13, h1, xw);
  k_scatter<<<(edgeThreads + 255) / 256, b256, 0, stream>>>(ei, ea, xw, agg);
  k_aggrelu<<<(nodeThreads + 255) / 256, b256, 0, stream>>>(agg, bc1, h1);

  // ---- Block 2 ----
  hipMemsetAsync(agg, 0, (size_t)N_NODES * 64 * sizeof(float), stream);
  k_node_gemm<NIN><<<gBlocks, b256, lds2, stream>>>(h1, W21, b21, Wc2, W22, b22,
                                                    W23, b23, h2, xw);
  k_scatter<<<(edgeThreads + 255) / 256, b256, 0, stream>>>(ei, ea, xw, agg);
  k_aggrelu<<<(nodeThreads + 255) / 256, b256, 0, stream>>>(agg, bc2, h2);

  // ---- Classifier + log_softmax ----
  k_classifier<<<gBlocks, b256, ldsC, stream>>>(h2, Wf, bf, out);
  k_logsoftmax<<<N_NODES, dim3(N_CLASSES), 0, stream>>>(out);
}